// Net_89773406421080
// MI455X (gfx1250) — compile-verified
//
#include <hip/hip_runtime.h>
#include <math.h>

typedef __attribute__((ext_vector_type(2))) float v2f;
typedef __attribute__((ext_vector_type(8))) float v8f;

// ---------------------------------------------------------------------------
// Zero a float buffer.
// ---------------------------------------------------------------------------
__global__ void sage_zero(float* __restrict__ p, long long n) {
    long long gid = (long long)blockIdx.x * blockDim.x + threadIdx.x;
    if (gid < n) p[gid] = 0.0f;
}

// ---------------------------------------------------------------------------
// Copy W[N,K] into Wp[Npad,K], zero-filling rows >= N (tile-align weights so
// the GEMM inner loop needs no bound checks).
// ---------------------------------------------------------------------------
__global__ void sage_pad_w(const float* __restrict__ W, float* __restrict__ Wp,
                           int N, int K, int Npad) {
    int idx = blockIdx.x * blockDim.x + threadIdx.x;
    int total = Npad * K;
    if (idx >= total) return;
    int row = idx / K;
    Wp[idx] = (row < N) ? W[idx] : 0.0f;
}

// ---------------------------------------------------------------------------
// Per-destination degree (float counts, as in reference).
// ---------------------------------------------------------------------------
__global__ void sage_degree(const long long* __restrict__ ei, long long E,
                            float* __restrict__ cnt) {
    long long e = (long long)blockIdx.x * blockDim.x + threadIdx.x;
    if (e >= E) return;
    long long dst = ei[E + e];
    atomicAdd(&cnt[dst], 1.0f);
}

// ---------------------------------------------------------------------------
// Dual GEMM via fp32 WMMA:  Cl = A @ Bl^T,  Cr = A @ Br^T
//   A      : [M, K] row-major
//   Bl/Br  : [Npad, K] row-major, Npad multiple of 32, rows >= N are zero
//   Cl/Cr  : [M, N] row-major (stores guarded by col < N)
// Each wave computes a 16x32 tile of BOTH outputs: one A fragment feeds
// 4 WMMAs. Hot loop: unconditional global_load_b64 only, software-pipelined
// one K-step ahead so loads stay in flight across the WMMA chain.
// V_WMMA_F32_16X16X4_F32 fragment layout (ISA 7.12.2):
//   A 16x4 : lanes 0-15 -> M=lane&15, VGPR v -> K=k+v ; lanes 16-31 -> K=k+2+v
//   B 4x16 : lanes 0-15 -> N=lane&15, VGPR v -> K=k+v ; lanes 16-31 -> K=k+2+v
//   C 16x16: VGPR v, lanes 0-15 -> (M=v, N=lane); lanes 16-31 -> (M=v+8, N=lane-16)
// ---------------------------------------------------------------------------
__global__ void sage_gemm_dual_wmma(const float* __restrict__ A,
                                    const float* __restrict__ Bl,
                                    const float* __restrict__ Br,
                                    float* __restrict__ Cl,
                                    float* __restrict__ Cr,
                                    int M, int K, int N,
                                    int pairTiles, int totalUnits) {
    const int wave = threadIdx.x >> 5;
    const int lane = threadIdx.x & 31;
    const int unit = blockIdx.x * (blockDim.x >> 5) + wave;
    if (unit >= totalUnits) return;          // wave-uniform exit: EXEC stays all-1s
    const int mTile = unit / pairTiles;
    const int pIdx  = unit - mTile * pairTiles;
    const int m0 = mTile * 16;
    const int n0 = pIdx * 32;

    const int r    = lane & 15;
    const int half = lane >> 4;              // 0: K pair {k,k+1}, 1: {k+2,k+3}

    int m = m0 + r;
    if (m >= M) m = M - 1;                   // clamp: load valid row, store guarded

    const float* aP  = A  + (size_t)m * K + 2 * half;
    const float* b0P = Bl + (size_t)(n0 + r)      * K + 2 * half;
    const float* b1P = Bl + (size_t)(n0 + 16 + r) * K + 2 * half;
    const float* c0P = Br + (size_t)(n0 + r)      * K + 2 * half;
    const float* c1P = Br + (size_t)(n0 + 16 + r) * K + 2 * half;

    v8f accL0 = {}, accL1 = {}, accR0 = {}, accR1 = {};

    // Prologue: load K-step 0.
    v2f a  = *(const v2f*)(aP);
    v2f b0 = *(const v2f*)(b0P);
    v2f b1 = *(const v2f*)(b1P);
    v2f c0 = *(const v2f*)(c0P);
    v2f c1 = *(const v2f*)(c1P);

#pragma unroll 4
    for (int k = 0; k < K; k += 4) {
        // Issue next K-step's loads before consuming the current fragments
        // (uniform clamp on last trip: redundant valid load, result unused).
        const int kn = (k + 4 < K) ? (k + 4) : k;
        v2f an  = *(const v2f*)(aP  + kn);
        v2f b0n = *(const v2f*)(b0P + kn);
        v2f b1n = *(const v2f*)(b1P + kn);
        v2f c0n = *(const v2f*)(c0P + kn);
        v2f c1n = *(const v2f*)(c1P + kn);

        accL0 = __builtin_amdgcn_wmma_f32_16x16x4_f32(false, a, false, b0,
                                                      (short)0, accL0, false, false);
        accL1 = __builtin_amdgcn_wmma_f32_16x16x4_f32(false, a, false, b1,
                                                      (short)0, accL1, false, false);
        accR0 = __builtin_amdgcn_wmma_f32_16x16x4_f32(false, a, false, c0,
                                                      (short)0, accR0, false, false);
        accR1 = __builtin_amdgcn_wmma_f32_16x16x4_f32(false, a, false, c1,
                                                      (short)0, accR1, false, false);

        a = an; b0 = b0n; b1 = b1n; c0 = c0n; c1 = c1n;
    }

    const int col0 = n0 + r;
    const int col1 = col0 + 16;
#pragma unroll
    for (int v = 0; v < 8; ++v) {
        const int row = m0 + v + half * 8;
        if (row < M) {
            float* cl = Cl + (size_t)row * N;
            float* cr = Cr + (size_t)row * N;
            if (col0 < N) { cl[col0] = accL0[v]; cr[col0] = accR0[v]; }
            if (col1 < N) { cl[col1] = accL1[v]; cr[col1] = accR1[v]; }
        }
    }
}

// ---------------------------------------------------------------------------
// Edge scatter-add, 128 floats per edge. One wave per edge, float4 per lane.
// Gathered Y (51 MB) is L2-resident on MI455X (192 MB L2).
// ---------------------------------------------------------------------------
__global__ void sage_scatter128(const long long* __restrict__ ei, long long E,
                                const float* __restrict__ Y,
                                float* __restrict__ Agg) {
    long long gid = (long long)blockIdx.x * blockDim.x + threadIdx.x;
    long long e = gid >> 5;
    int lane = (int)(gid & 31);
    if (e >= E) return;
    long long src = ei[e];
    long long dst = ei[E + e];
    const float4 v = *(const float4*)(Y + src * 128 + lane * 4);
    float* dp = Agg + dst * 128 + lane * 4;
    atomicAdd(dp + 0, v.x);
    atomicAdd(dp + 1, v.y);
    atomicAdd(dp + 2, v.z);
    atomicAdd(dp + 3, v.w);
}

// ---------------------------------------------------------------------------
// h = relu(agg/max(cnt,1) + bias + self), hid = 128
// ---------------------------------------------------------------------------
__global__ void sage_combine_relu(const float* __restrict__ Agg,
                                  const float* __restrict__ cnt,
                                  const float* __restrict__ bias,
                                  const float* __restrict__ Self,
                                  float* __restrict__ H, long long total) {
    long long gid = (long long)blockIdx.x * blockDim.x + threadIdx.x;
    if (gid >= total) return;
    long long i = gid >> 7;         // / 128
    int c = (int)(gid & 127);
    float d = cnt[i];
    d = d > 1.0f ? d : 1.0f;
    float v = Agg[gid] / d + bias[c] + Self[gid];
    H[gid] = v > 0.0f ? v : 0.0f;
}

// ---------------------------------------------------------------------------
// Edge scatter-add, 40 floats per edge (layer-2, post-GEMM thanks to
// aggregation/linear commutativity).
// ---------------------------------------------------------------------------
__global__ void sage_scatter40(const long long* __restrict__ ei, long long E,
                               const float* __restrict__ Z,
                               float* __restrict__ Agg) {
    long long gid = (long long)blockIdx.x * blockDim.x + threadIdx.x;
    if (gid >= E * 40) return;
    long long e = gid / 40;
    int c = (int)(gid - e * 40);
    long long src = ei[e];
    long long dst = ei[E + e];
    atomicAdd(&Agg[dst * 40 + c], Z[src * 40 + c]);
}

// ---------------------------------------------------------------------------
// Final: o = agg/max(cnt,1) + bias + self ; out = log_softmax(o), 40 classes.
// One thread per node; 40 values live in registers.
// ---------------------------------------------------------------------------
__global__ void sage_final_logsoftmax(const float* __restrict__ Agg,
                                      const float* __restrict__ cnt,
                                      const float* __restrict__ bias,
                                      const float* __restrict__ Self,
                                      float* __restrict__ Out, int M) {
    int i = blockIdx.x * blockDim.x + threadIdx.x;
    if (i >= M) return;
    float d = cnt[i];
    d = d > 1.0f ? d : 1.0f;
    const float* ag = Agg + (size_t)i * 40;
    const float* sf = Self + (size_t)i * 40;
    float o[40];
    float mx = -INFINITY;
#pragma unroll
    for (int c = 0; c < 40; ++c) {
        float v = ag[c] / d + bias[c] + sf[c];
        o[c] = v;
        mx = v > mx ? v : mx;
    }
    float s = 0.0f;
#pragma unroll
    for (int c = 0; c < 40; ++c) s += expf(o[c] - mx);
    float lse = logf(s) + mx;
    float* op = Out + (size_t)i * 40;
#pragma unroll
    for (int c = 0; c < 40; ++c) op[c] = o[c] - lse;
}

// ---------------------------------------------------------------------------
extern "C" void kernel_launch(void* const* d_in, const int* in_sizes, int n_in,
                              void* d_out, int out_size, void* d_ws, size_t ws_size,
                              hipStream_t stream) {
    const float*     x   = (const float*)d_in[0];
    const long long* ei  = (const long long*)d_in[1];
    const float*     Wl1 = (const float*)d_in[2];
    const float*     bl1 = (const float*)d_in[3];
    const float*     Wr1 = (const float*)d_in[4];
    const float*     Wl2 = (const float*)d_in[5];
    const float*     bl2 = (const float*)d_in[6];
    const float*     Wr2 = (const float*)d_in[7];
    float* out = (float*)d_out;

    const int IN_C = 256, HID = 128, OUTC = 40, OUTP = 64;  // OUTP: padded, mult of 32
    const long long M = in_sizes[0] / IN_C;   // 100000
    const long long E = in_sizes[1] / 2;      // 1600000

    // Workspace layout (floats):
    //   cnt  [M]         degree
    //   bufC [M*HID]     agg1  -> later: zr [M*40] | agg2 [M*40]
    //   bufA [M*HID]     yl = x@Wl1^T -> later h
    //   bufB [M*HID]     yr = x@Wr1^T -> later zl = h@Wl2^T
    //   Wl2p [OUTP*HID]  zero-padded layer-2 weights
    //   Wr2p [OUTP*HID]
    float* ws   = (float*)d_ws;
    float* cnt  = ws;
    float* bufC = cnt + M;
    float* bufA = bufC + (size_t)M * HID;
    float* bufB = bufA + (size_t)M * HID;
    float* Wl2p = bufB + (size_t)M * HID;
    float* Wr2p = Wl2p + (size_t)OUTP * HID;
    float* zr   = bufC;
    float* agg2 = bufC + (size_t)M * OUTC;

    const int BS = 256;
    const int wavesPerBlock = BS / 32;

    // 0) pad layer-2 weights to 64x128 (tile-aligned, zero-filled)
    {
        int n = OUTP * HID;
        sage_pad_w<<<(n + BS - 1) / BS, BS, 0, stream>>>(Wl2, Wl2p, OUTC, HID, OUTP);
        sage_pad_w<<<(n + BS - 1) / BS, BS, 0, stream>>>(Wr2, Wr2p, OUTC, HID, OUTP);
    }
    // 1) zero cnt + agg1 (contiguous region)
    {
        long long n = M + M * (long long)HID;
        sage_zero<<<(int)((n + BS - 1) / BS), BS, 0, stream>>>(cnt, n);
    }
    // 2) degrees
    sage_degree<<<(int)((E + BS - 1) / BS), BS, 0, stream>>>(ei, E, cnt);

    // 3) layer-1 dual GEMM: yl = x@Wl1^T, yr = x@Wr1^T   (K=256, N=Npad=128)
    {
        int mTiles = (int)((M + 15) / 16);
        int pairTiles = HID / 32;                     // 4
        int total = mTiles * pairTiles;
        int blocks = (total + wavesPerBlock - 1) / wavesPerBlock;
        sage_gemm_dual_wmma<<<blocks, BS, 0, stream>>>(x, Wl1, Wr1, bufA, bufB,
                                                       (int)M, IN_C, HID,
                                                       pairTiles, total);
    }
    // 4) scatter yl over edges into agg1
    {
        long long thr = E * 32;
        sage_scatter128<<<(int)((thr + BS - 1) / BS), BS, 0, stream>>>(ei, E, bufA, bufC);
    }
    // 5) h = relu(agg1/cnt + bl1 + yr)  (overwrites bufA)
    {
        long long n = M * (long long)HID;
        sage_combine_relu<<<(int)((n + BS - 1) / BS), BS, 0, stream>>>(bufC, cnt, bl1,
                                                                       bufB, bufA, n);
    }
    // 6) layer-2 dual GEMM: zl = h@Wl2^T (bufB), zr = h@Wr2^T (bufC)  (K=128, N=40, pad 64)
    {
        int mTiles = (int)((M + 15) / 16);
        int pairTiles = OUTP / 32;                    // 2
        int total = mTiles * pairTiles;
        int blocks = (total + wavesPerBlock - 1) / wavesPerBlock;
        sage_gemm_dual_wmma<<<blocks, BS, 0, stream>>>(bufA, Wl2p, Wr2p, bufB, zr,
                                                       (int)M, HID, OUTC,
                                                       pairTiles, total);
    }
    // 7) zero agg2, scatter zl over edges
    {
        long long n = M * (long long)OUTC;
        sage_zero<<<(int)((n + BS - 1) / BS), BS, 0, stream>>>(agg2, n);
        long long thr = E * (long long)OUTC;
        sage_scatter40<<<(int)((thr + BS - 1) / BS), BS, 0, stream>>>(ei, E, bufB, agg2);
    }
    // 8) final combine + log_softmax
    sage_final_logsoftmax<<<(int)((M + BS - 1) / BS), BS, 0, stream>>>(agg2, cnt, bl2,
                                                                      zr, out, (int)M);
}